// RepulsionLoss_26414048871077
// MI455X (gfx1250) — compile-verified
//
#include <hip/hip_runtime.h>
#include <hip/hip_bf16.h>

// RepulsionLoss for MI455X (gfx1250, wave32).
// d_in[0] = loc_data   [B,N,4] f32
// d_in[1] = ground_data[B,G,4] f32
// d_in[2] = prior_data [N,4]   f32
// d_in[3] = pos_idx    [B,N,4] bool (1 byte per element, numpy bool_ layout)
// d_out   = scalar f32
// d_ws    = pred boxes [B,N,4] f32 + per-block (sum,count) partials.

#define TPB 256
#define WAVES (TPB / 32)

typedef float v2f __attribute__((ext_vector_type(2)));
typedef float v8f __attribute__((ext_vector_type(8)));

// ---------------------------------------------------------------------------
// Deterministic block-wide (sum, count) reduction.
// Wave level uses V_WMMA_F32_16X16X4_F32:
//   A (16x4): lane m (m<16) supplies A[m][0]=s, A[m][1]=c; lane m+16 supplies
//   A[m][2]=s, A[m][3]=c.  B (4x16) is a selector with B[0][0]=B[2][0]=1 and
//   B[1][1]=B[3][1]=1, so D[m][0] = s_m + s_{m+16}, D[m][1] = c_m + c_{m+16}.
//   D layout: VGPR v holds M=v (lanes 0-15, N=lane) and M=v+8 (lanes 16-31).
//   => lane0 + lane16 own all 16 row-sums of s; lane1 + lane17 own c.
// Cross-wave combine is a fixed-order scalar loop (deterministic).
// Must be called by all 256 threads (EXEC all ones for WMMA).
// ---------------------------------------------------------------------------
__device__ __forceinline__ float2 block_reduce_pair(float s, float c) {
    __shared__ float part[WAVES][4];
    const int tid  = threadIdx.x;
    const int wave = tid >> 5;
    const int lane = tid & 31;

    v2f a; a[0] = s; a[1] = c;
    v2f b;
    b[0] = ((lane & 15) == 0) ? 1.0f : 0.0f;   // rows K=0 (lanes<16) / K=2 (lanes>=16), col 0
    b[1] = ((lane & 15) == 1) ? 1.0f : 0.0f;   // rows K=1 / K=3, col 1
    v8f acc = {0.f, 0.f, 0.f, 0.f, 0.f, 0.f, 0.f, 0.f};
    acc = __builtin_amdgcn_wmma_f32_16x16x4_f32(false, a, false, b, (short)0, acc,
                                                false, false);
    float local = ((acc[0] + acc[1]) + (acc[2] + acc[3])) +
                  ((acc[4] + acc[5]) + (acc[6] + acc[7]));
    if (lane == 0)  part[wave][0] = local;   // sums, rows 0..7
    if (lane == 16) part[wave][1] = local;   // sums, rows 8..15
    if (lane == 1)  part[wave][2] = local;   // counts, rows 0..7
    if (lane == 17) part[wave][3] = local;   // counts, rows 8..15
    __syncthreads();
    if (tid == 0) {
        float S = 0.f, C = 0.f;
        for (int w = 0; w < WAVES; ++w) {
            S += part[w][0] + part[w][1];
            C += part[w][2] + part[w][3];
        }
        part[0][0] = S; part[0][1] = C;
    }
    __syncthreads();
    float2 r; r.x = part[0][0]; r.y = part[0][1];
    __syncthreads();
    return r;
}

// ---------------------------------------------------------------------------
// Kernel 1: SSD decode  (cx,cy,w,h priors + offsets -> corner boxes)
// ---------------------------------------------------------------------------
__global__ void k_decode(const float* __restrict__ loc, const float* __restrict__ prior,
                         float* __restrict__ boxes, int B, int N) {
    int t = blockIdx.x * blockDim.x + threadIdx.x;
    if (t >= B * N) return;
    int i = t % N;
    const float* l = loc   + (size_t)t * 4;
    const float* p = prior + (size_t)i * 4;
    float cx = p[0] + l[0] * 0.1f * p[2];
    float cy = p[1] + l[1] * 0.1f * p[3];
    float w  = p[2] * expf(l[2] * 0.2f);
    float h  = p[3] * expf(l[3] * 0.2f);
    float* o = boxes + (size_t)t * 4;
    o[0] = cx - w * 0.5f; o[1] = cy - h * 0.5f;
    o[2] = cx + w * 0.5f; o[3] = cy + h * 0.5f;
}

__device__ __forceinline__ float iou_corner(float ax0, float ay0, float ax1, float ay1,
                                            float areaA,
                                            float bx0, float by0, float bx1, float by1) {
    float lx = fmaxf(ax0, bx0), ly = fmaxf(ay0, by0);
    float rx = fminf(ax1, bx1), ry = fminf(ay1, by1);
    float w = fmaxf(rx - lx, 0.f), h = fmaxf(ry - ly, 0.f);
    float inter = w * h;
    float areaB = (bx1 - bx0) * (by1 - by0);
    return inter / (areaA + areaB - inter);
}

// ---------------------------------------------------------------------------
// Kernel 2: repgt partials.  grid = B*chunks blocks, 256 thr.
// Replicates jnp.argmax first-occurrence tie-breaking exactly.
// ---------------------------------------------------------------------------
__global__ void k_repgt(const float* __restrict__ boxes, const float* __restrict__ gt,
                        const unsigned char* __restrict__ pos,
                        float* __restrict__ partials, int B, int N, int G, int chunks) {
    __shared__ float sg[512];                       // up to G=128 boxes
    int blk = blockIdx.x;
    int b = blk / chunks, chunk = blk % chunks;
    int i = chunk * TPB + threadIdx.x;
    for (int t = threadIdx.x; t < G * 4; t += TPB) sg[t] = gt[(size_t)b * G * 4 + t];
    __syncthreads();

    float term = 0.f, cnt = 0.f;
    if (i < N && pos[((size_t)b * N + i) * 4] != 0) {
        const float* bx = boxes + ((size_t)b * N + i) * 4;
        float ax0 = bx[0], ay0 = bx[1], ax1 = bx[2], ay1 = bx[3];
        float areaA = (ax1 - ax0) * (ay1 - ay0);
        // pass 1: argmax (first occurrence; all ov >= 0 so init -1 always loses)
        float max1 = -1.f; int arg1 = 0;
        for (int j = 0; j < G; ++j) {
            const float* g = &sg[j * 4];
            float ov = iou_corner(ax0, ay0, ax1, ay1, areaA, g[0], g[1], g[2], g[3]);
            if (ov > max1) { max1 = ov; arg1 = j; }
        }
        // pass 2: argmax of ov with column arg1 zeroed (scan all columns)
        float max2 = -1.f; int arg2 = 0;
        for (int j = 0; j < G; ++j) {
            const float* g = &sg[j * 4];
            float ov = iou_corner(ax0, ay0, ax1, ay1, areaA, g[0], g[1], g[2], g[3]);
            float o2 = (j == arg1) ? 0.f : ov;
            if (o2 > max2) { max2 = o2; arg2 = j; }
        }
        if (max2 > 0.f) {
            const float* g = &sg[arg2 * 4];
            float lx = fmaxf(ax0, g[0]), ly = fmaxf(ay0, g[1]);
            float rx = fminf(ax1, g[2]), ry = fminf(ay1, g[3]);
            float w = fmaxf(rx - lx, 0.f), h = fmaxf(ry - ly, 0.f);
            float inter  = w * h;
            float g_area = (g[2] - g[0]) * (g[3] - g[1]);
            float iog = inter / g_area;
            const float s = 0.9f;
            float t2;
            if (iog > s) {
                t2 = (iog - s) / (1.0f - s) + 2.3025851f;   // -log(1-0.9)
            } else {
                float v = 1.0f - iog;
                if (v < 1e-10f) v = 1e-10f;
                t2 = -logf(v);
            }
            term = t2; cnt = 1.f;
        }
    }
    float2 r = block_reduce_pair(term, cnt);
    if (threadIdx.x == 0) { partials[blk * 2] = r.x; partials[blk * 2 + 1] = r.y; }
}

// ---------------------------------------------------------------------------
// Kernel 3: repbox partials.  sigma=0 => term == ov for ov>0.
// Thread owns row i; j-tiles of 256 boxes (+mask) staged through LDS.
// ---------------------------------------------------------------------------
__global__ void k_repbox(const float* __restrict__ boxes,
                         const unsigned char* __restrict__ pos,
                         float* __restrict__ partials, int B, int N, int chunks) {
    __shared__ float tb[TPB * 4];
    __shared__ float tm[TPB];
    int blk = blockIdx.x;
    int b = blk / chunks, chunk = blk % chunks;
    int i = chunk * TPB + threadIdx.x;

    float ax0 = 0.f, ay0 = 0.f, ax1 = 0.f, ay1 = 0.f, areaA = 0.f;
    bool act = false;
    if (i < N) {
        const float* bx = boxes + ((size_t)b * N + i) * 4;
        ax0 = bx[0]; ay0 = bx[1]; ax1 = bx[2]; ay1 = bx[3];
        areaA = (ax1 - ax0) * (ay1 - ay0);
        act = pos[((size_t)b * N + i) * 4] != 0;
    }

    float sum = 0.f, cnt = 0.f;
    for (int j0 = 0; j0 < N; j0 += TPB) {
        __syncthreads();
        int j = j0 + threadIdx.x;
        if (j < N) {
            const float* bj = boxes + ((size_t)b * N + j) * 4;
            tb[threadIdx.x * 4 + 0] = bj[0];
            tb[threadIdx.x * 4 + 1] = bj[1];
            tb[threadIdx.x * 4 + 2] = bj[2];
            tb[threadIdx.x * 4 + 3] = bj[3];
            tm[threadIdx.x] = pos[((size_t)b * N + j) * 4] ? 1.f : 0.f;
        }
        // hint next tile into cache (global_prefetch_b8)
        if (j + TPB < N)
            __builtin_prefetch(boxes + ((size_t)b * N + j + TPB) * 4, 0, 0);
        __syncthreads();
        if (act) {
            int lim = min(TPB, N - j0);
            for (int jj = 0; jj < lim; ++jj) {
                if (tm[jj] > 0.f) {
                    float ov = iou_corner(ax0, ay0, ax1, ay1, areaA,
                                          tb[jj * 4 + 0], tb[jj * 4 + 1],
                                          tb[jj * 4 + 2], tb[jj * 4 + 3]);
                    if (ov > 0.f) { sum += ov; cnt += 1.f; }
                }
            }
        }
    }
    float2 r = block_reduce_pair(sum, cnt);
    if (threadIdx.x == 0) { partials[blk * 2] = r.x; partials[blk * 2 + 1] = r.y; }
}

// ---------------------------------------------------------------------------
// Kernel 4: deterministic serial combine -> scalar loss
// ---------------------------------------------------------------------------
__global__ void k_final(const float* __restrict__ gtP, const float* __restrict__ boxP,
                        float* __restrict__ out, int B, int chunks) {
    if (blockIdx.x == 0 && threadIdx.x == 0) {
        float loss = 0.f;
        for (int b = 0; b < B; ++b) {
            float s = 0.f, c = 0.f, s2 = 0.f, c2 = 0.f;
            for (int k = 0; k < chunks; ++k) {
                s  += gtP[(b * chunks + k) * 2];
                c  += gtP[(b * chunks + k) * 2 + 1];
                s2 += boxP[(b * chunks + k) * 2];
                c2 += boxP[(b * chunks + k) * 2 + 1];
            }
            if (c  > 0.f) loss += s  / fmaxf(c,  1.f);
            if (c2 > 0.f) loss += s2 / fmaxf(c2, 1.f);
        }
        out[0] = loss;
    }
}

extern "C" void kernel_launch(void* const* d_in, const int* in_sizes, int n_in,
                              void* d_out, int out_size, void* d_ws, size_t ws_size,
                              hipStream_t stream) {
    const float* loc   = (const float*)d_in[0];
    const float* gt    = (const float*)d_in[1];
    const float* prior = (const float*)d_in[2];
    const unsigned char* pos = (const unsigned char*)d_in[3];
    float* out = (float*)d_out;

    const int N = in_sizes[2] / 4;              // prior_data [N,4]
    const int B = in_sizes[0] / (4 * N);        // loc_data   [B,N,4]
    const int G = in_sizes[1] / (4 * B);        // ground     [B,G,4]
    const int chunks = (N + TPB - 1) / TPB;

    float* W      = (float*)d_ws;
    float* boxesW = W;                          // B*N*4 floats
    float* gtP    = boxesW + (size_t)B * N * 4; // 2*B*chunks floats
    float* boxP   = gtP    + (size_t)2 * B * chunks;

    int total = B * N;
    k_decode<<<(total + TPB - 1) / TPB, TPB, 0, stream>>>(loc, prior, boxesW, B, N);
    k_repgt <<<B * chunks, TPB, 0, stream>>>(boxesW, gt, pos, gtP, B, N, G, chunks);
    k_repbox<<<B * chunks, TPB, 0, stream>>>(boxesW, pos, boxP, B, N, chunks);
    k_final <<<1, 32, 0, stream>>>(gtP, boxP, out, B, chunks);
}